// QuantumNeuralNetwork_12524124635253
// MI455X (gfx1250) — compile-verified
//
#include <hip/hip_runtime.h>
#include <hip/hip_bf16.h>
#include <math.h>

// ---------------------------------------------------------------------------
// Quantum NN forward:
//   angles = tanh(x @ W_pre.T + b_pre)                 [32768, 10]
//   q_out  = 10-qubit statevector sim (RX embed + 4x StronglyEntangling)
//   out    = sigmoid(q_out @ W_post.T + b_post)        [32768, 1]
// ---------------------------------------------------------------------------

#define BATCH     32768
#define NFEAT     128
#define NQ        10
#define QDEPTH    4
#define DIM       1024          // 2^NQ
#define WAVES_PB  8             // kernel-2 waves per block (8 * 8KB LDS = 64KB)

typedef __attribute__((ext_vector_type(2))) float v2f;
typedef __attribute__((ext_vector_type(8))) float v8f;

// ===========================================================================
// Kernel 1: pre-net GEMM via V_WMMA_F32_16X16X4_F32 + bias + tanh.
// One wave32 computes a 16(M=samples) x 16(N=outputs, 10 valid) tile,
// stepping K = 128 in 32 WMMA steps of K=4.
//
// A (16x4 f32) VGPR layout: lane L holds M=L%16, K-pair = (L/16)*2 .. +1
// B (4x16 f32) VGPR layout: lane L holds N=L%16, K-pair = (L/16)*2 .. +1
// C (16x16 f32): VGPR j -> M = j + (L/16)*8, N = L%16
//
// No exec-mask games in the hot loop: lanes with n >= 10 load a clamped
// (in-bounds) W_pre row and multiply the fragment by a 0/1 mask, keeping
// the load stream branch-free (WMMA requires EXEC all-ones anyway).
// ===========================================================================
__global__ __launch_bounds__(128)
void qnn_prenet_wmma(const float* __restrict__ x,      // [BATCH, NFEAT]
                     const float* __restrict__ Wpre,   // [NQ, NFEAT]
                     const float* __restrict__ bpre,   // [NQ]
                     float* __restrict__ angles)       // [BATCH, NQ]
{
    const int lane   = threadIdx.x & 31;
    const int wave   = threadIdx.x >> 5;
    const int gwave  = blockIdx.x * 4 + wave;          // 4 waves / block
    const int tile_m = gwave * 16;                     // 16 samples per tile

    const int   n     = lane & 15;                     // output column
    const int   kh    = (lane >> 4) << 1;              // K sub-pair: 0 or 2
    const int   nclmp = (n < NQ) ? n : (NQ - 1);       // in-bounds row
    const float bmask = (n < NQ) ? 1.f : 0.f;          // zero-pad columns >= 10

    const int row = tile_m + (lane & 15);

    const float* __restrict__ arow = x    + (size_t)row   * NFEAT + kh;
    const float* __restrict__ wrow = Wpre + (size_t)nclmp * NFEAT + kh;

    v8f c = {0.f, 0.f, 0.f, 0.f, 0.f, 0.f, 0.f, 0.f};

    #pragma unroll
    for (int kb = 0; kb < NFEAT; kb += 4) {
        v2f a = *(const v2f*)(arow + kb);      // global_load_b64
        v2f b = *(const v2f*)(wrow + kb);      // global_load_b64 (unconditional)
        b.x *= bmask;
        b.y *= bmask;
        c = __builtin_amdgcn_wmma_f32_16x16x4_f32(
                /*neg_a=*/false, a, /*neg_b=*/false, b,
                /*c_mod=*/(short)0, c, /*reuse_a=*/false, /*reuse_b=*/false);
    }

    // Epilogue: bias + tanh, scatter valid columns (n < 10)
    if (n < NQ) {
        const float bias   = bpre[n];
        const int   m_base = (lane >> 4) * 8;
        #pragma unroll
        for (int j = 0; j < 8; ++j) {
            const int m = m_base + j;
            angles[(size_t)(tile_m + m) * NQ + n] = tanhf(c[j] + bias);
        }
    }
}

// ===========================================================================
// Kernel 2: statevector simulation, one wave32 per sample, state in LDS.
// ===========================================================================

// Apply uniform 2x2 complex gate U on bit position p (0 = LSB of state index).
__device__ __forceinline__
void apply1q(float2* __restrict__ psi, int p,
             float u00r, float u00i, float u01r, float u01i,
             float u10r, float u10i, float u11r, float u11i, int lane)
{
    const int mask = 1 << p;
    #pragma unroll 4
    for (int pr = lane; pr < (DIM / 2); pr += 32) {
        const int low = pr & (mask - 1);
        const int k0  = ((pr - low) << 1) | low;   // insert 0 at bit p
        const int k1  = k0 | mask;
        const float2 a0 = psi[k0];
        const float2 a1 = psi[k1];
        float2 r0, r1;
        r0.x = u00r * a0.x - u00i * a0.y + u01r * a1.x - u01i * a1.y;
        r0.y = u00r * a0.y + u00i * a0.x + u01r * a1.y + u01i * a1.x;
        r1.x = u10r * a0.x - u10i * a0.y + u11r * a1.x - u11i * a1.y;
        r1.y = u10r * a0.y + u10i * a0.x + u11r * a1.y + u11i * a1.x;
        psi[k0] = r0;
        psi[k1] = r1;
    }
}

// CNOT: control bit pc, target bit pt (state-index bit positions, pc != pt).
__device__ __forceinline__
void cnot(float2* __restrict__ psi, int pc, int pt, int lane)
{
    const int tmask = 1 << pt;
    #pragma unroll 4
    for (int pr = lane; pr < (DIM / 2); pr += 32) {
        const int low = pr & (tmask - 1);
        const int k0  = ((pr - low) << 1) | low;
        if ((k0 >> pc) & 1) {
            const int k1 = k0 | tmask;
            const float2 t0 = psi[k0];
            const float2 t1 = psi[k1];
            psi[k0] = t1;
            psi[k1] = t0;
        }
    }
}

__global__ __launch_bounds__(WAVES_PB * 32)
void qnn_sim(const float* __restrict__ angles,   // [BATCH, NQ]
             const float* __restrict__ qw,       // [QDEPTH, NQ, 3]
             const float* __restrict__ Wpost,    // [1, NQ]
             const float* __restrict__ bpost,    // [1]
             float* __restrict__ out)            // [BATCH, 1]
{
    __shared__ float2 sv[WAVES_PB * DIM];        // 64 KB

    const int wave = threadIdx.x >> 5;
    const int lane = threadIdx.x & 31;
    const int b    = blockIdx.x * WAVES_PB + wave;

    float2* psi = &sv[wave * DIM];

    // |0...0>
    for (int k = lane; k < DIM; k += 32)
        psi[k] = make_float2(k == 0 ? 1.f : 0.f, 0.f);
    __syncthreads();

    // AngleEmbedding: RX(angles[b, w]) on wire w; wire w <-> bit (NQ-1-w)
    for (int w = 0; w < NQ; ++w) {
        float st, ct;
        sincosf(0.5f * angles[(size_t)b * NQ + w], &st, &ct);
        // RX = [[c, -i s], [-i s, c]]
        apply1q(psi, NQ - 1 - w,
                ct, 0.f,   0.f, -st,
                0.f, -st,  ct, 0.f, lane);
        __syncthreads();
    }

    // StronglyEntanglingLayers
    for (int l = 0; l < QDEPTH; ++l) {
        for (int w = 0; w < NQ; ++w) {
            const float phi = qw[((size_t)l * NQ + w) * 3 + 0];
            const float th  = qw[((size_t)l * NQ + w) * 3 + 1];
            const float om  = qw[((size_t)l * NQ + w) * 3 + 2];
            float st, ct, sap, cap, sam, cam;
            sincosf(0.5f * th,         &st,  &ct);
            sincosf(0.5f * (phi + om), &sap, &cap);
            sincosf(0.5f * (phi - om), &sam, &cam);
            // Rot = RZ(om) RY(th) RZ(phi):
            //  u00 = e^{-i ap} c;  u01 = -e^{+i am} s
            //  u10 = e^{-i am} s;  u11 = e^{+i ap} c
            apply1q(psi, NQ - 1 - w,
                    cap * ct, -sap * ct,   -cam * st, -sam * st,
                    cam * st, -sam * st,    cap * ct,  sap * ct, lane);
            __syncthreads();
        }
        const int r = (l % (NQ - 1)) + 1;
        for (int w = 0; w < NQ; ++w) {
            const int tw = (w + r) % NQ;
            cnot(psi, NQ - 1 - w, NQ - 1 - tw, lane);
            __syncthreads();
        }
    }

    // <Z_i> = sum_k |psi_k|^2 * (bit_{NQ-1-i}(k) ? -1 : +1)
    float ez[NQ];
    #pragma unroll
    for (int i = 0; i < NQ; ++i) ez[i] = 0.f;

    for (int k = lane; k < DIM; k += 32) {
        const float2 a = psi[k];
        const float p  = a.x * a.x + a.y * a.y;
        #pragma unroll
        for (int i = 0; i < NQ; ++i)
            ez[i] += ((k >> (NQ - 1 - i)) & 1) ? -p : p;
    }

    // wave32 tree reduction
    #pragma unroll
    for (int off = 16; off > 0; off >>= 1) {
        #pragma unroll
        for (int i = 0; i < NQ; ++i)
            ez[i] += __shfl_xor(ez[i], off, 32);
    }

    if (lane == 0) {
        float acc = bpost[0];
        #pragma unroll
        for (int i = 0; i < NQ; ++i)
            acc += ez[i] * Wpost[i];
        out[b] = 1.f / (1.f + expf(-acc));
    }
}

// ===========================================================================
// Launch
// ===========================================================================
extern "C" void kernel_launch(void* const* d_in, const int* in_sizes, int n_in,
                              void* d_out, int out_size, void* d_ws, size_t ws_size,
                              hipStream_t stream)
{
    const float* x     = (const float*)d_in[0];   // [BATCH, NFEAT]
    const float* Wpre  = (const float*)d_in[1];   // [NQ, NFEAT]
    const float* bpre  = (const float*)d_in[2];   // [NQ]
    const float* qw    = (const float*)d_in[3];   // [QDEPTH, NQ, 3]
    const float* Wpost = (const float*)d_in[4];   // [1, NQ]
    const float* bpost = (const float*)d_in[5];   // [1]
    float*       outp  = (float*)d_out;           // [BATCH, 1]
    float*       ang   = (float*)d_ws;            // [BATCH, NQ] scratch

    // Kernel 1: 2048 tiles of 16 rows, 4 waves/block (128 threads)
    {
        const int tiles  = BATCH / 16;            // 2048
        const int blocks = tiles / 4;             // 512
        qnn_prenet_wmma<<<blocks, 128, 0, stream>>>(x, Wpre, bpre, ang);
    }

    // Kernel 2: one wave32 per sample, 8 waves/block
    {
        const int blocks = BATCH / WAVES_PB;      // 4096
        qnn_sim<<<blocks, WAVES_PB * 32, 0, stream>>>(ang, qw, Wpost, bpost, outp);
    }
}